// ScalarBorn_50199577756038
// MI455X (gfx1250) — compile-verified
//
#include <hip/hip_runtime.h>

typedef float v2f  __attribute__((ext_vector_type(2)));
typedef float v8f  __attribute__((ext_vector_type(8)));
typedef unsigned int u32x4 __attribute__((ext_vector_type(4)));
typedef int   i32x4 __attribute__((ext_vector_type(4)));
typedef int   i32x8 __attribute__((ext_vector_type(8)));

#define NN      444            // padded grid edge (400 + 2*22)
#define NYX     (NN*NN)
#define NSH     4              // shots
#define NTT     300            // time steps
#define NRC     100            // receivers
#define PADC    22
#define DT_C    0.0005f
#define INV_H   0.2f           // 1/DX
#define INV_H2  0.04f          // 1/DX^2

// C1 applied at offsets -2,-1,+1,+2 (center coef is 0)
#define C1M2 (1.0f/12.0f)
#define C1M1 (-2.0f/3.0f)
#define C1P1 (2.0f/3.0f)
#define C1P2 (-1.0f/12.0f)

// ---------------- LDS tile container (TX=32, TY=16, halo 4) ----------------
struct Tiles {
    float sW[24][40];    // wavefield tile, halo 4 each side (MUST be first: TDM lds_addr=0)
    float sPY[20][32];   // psiy_new over rows [-2,17]
    float sPX[16][36];   // psix_new over cols [-2,33]
    float sD2X[16][32];  // d2/dx2 (WMMA)
    float sD2Y[16][32];  // d2/dy2 (WMMA)
};

// Banded 2nd-derivative coefficient (symmetric): branchless selects, no memory.
__device__ __forceinline__ float band_c2(int d) {
    float r = (d == 0 || d == 4) ? (-INV_H2 / 12.0f) : 0.0f;
    r = (d == 1 || d == 3) ? (INV_H2 * 4.0f / 3.0f) : r;
    r = (d == 2) ? (-INV_H2 * 2.5f) : r;
    return r;
}

// ---------------- WMMA second derivatives ----------------
// d2x tile h: out(16x16) = A(16x20)*B(20x16); A[m][k]=sW[m+4][h*16+2+k], B=banded (cf)
__device__ __forceinline__ void d2x_wmma(const float (&sW)[24][40], float (&sOut)[16][32],
                                         int h, int lane, const v2f (&cf)[5]) {
    const int half = lane >> 4, l = lane & 15;
    const int c0 = h*16 + 2 + 2*half;          // lanes>=16 hold K+2,K+3 of each chunk
    v2f afrag[5];
#pragma unroll
    for (int q = 0; q < 5; ++q) {              // batch LDS reads before the WMMA chain
        afrag[q].x = sW[l + 4][c0 + 4*q];
        afrag[q].y = sW[l + 4][c0 + 4*q + 1];
    }
    v8f acc = {0.f,0.f,0.f,0.f,0.f,0.f,0.f,0.f};
#pragma unroll
    for (int q = 0; q < 5; ++q)
        acc = __builtin_amdgcn_wmma_f32_16x16x4_f32(false, afrag[q], false, cf[q],
                                                    (short)0, acc, false, false);
#pragma unroll
    for (int r = 0; r < 8; ++r)                // C layout: VGPR r -> M=r / M=r+8
        sOut[r + 8*half][h*16 + l] = acc[r];
}

// d2y tile h: out = A(16x20)*B(20x16); A=banded (cf), B[k][c]=sW[k+2][h*16+c+4]
__device__ __forceinline__ void d2y_wmma(const float (&sW)[24][40], float (&sOut)[16][32],
                                         int h, int lane, const v2f (&cf)[5]) {
    const int half = lane >> 4, l = lane & 15;
    const int cc = h*16 + l + 4;
    const int r0 = 2 + 2*half;
    v2f bfrag[5];
#pragma unroll
    for (int q = 0; q < 5; ++q) {
        bfrag[q].x = sW[r0 + 4*q][cc];
        bfrag[q].y = sW[r0 + 4*q + 1][cc];
    }
    v8f acc = {0.f,0.f,0.f,0.f,0.f,0.f,0.f,0.f};
#pragma unroll
    for (int q = 0; q < 5; ++q)
        acc = __builtin_amdgcn_wmma_f32_16x16x4_f32(false, cf[q], false, bfrag[q],
                                                    (short)0, acc, false, false);
#pragma unroll
    for (int r = 0; r < 8; ++r)
        sOut[r + 8*half][h*16 + l] = acc[r];
}

// ---------------- tile load: TDM for interior blocks, loop for edges ----------------
__device__ __forceinline__ void load_tile(Tiles& T, const float* __restrict__ wfc,
                                          int X0, int Y0, int tid) {
#if __has_builtin(__builtin_amdgcn_tensor_load_to_lds) && __has_builtin(__builtin_amdgcn_s_wait_tensorcnt)
    const bool interior = (X0 >= 4) && (Y0 >= 4) && (X0 + 36 <= NN) && (Y0 + 20 <= NN);
    if (interior) {                              // block-uniform branch
        if (tid < 32) {                          // wave 0 issues one 2-D DMA: 24x40 f32 tile
            const unsigned long long ga =
                (unsigned long long)(const void*)(wfc + (size_t)(Y0 - 4)*NN + (X0 - 4));
            // D# group0: count=1 | lds_addr=0 (sW is at LDS offset 0) | global_addr | type=2
            u32x4 g0 = { 1u, 0u, (unsigned)ga,
                         (unsigned)((ga >> 32) & 0x1FFFFFFu) | (2u << 30) };
            // D# group1: data_size=4B; tensor_dim0/1=NN; tile=40x24; dim0_stride=NN
            i32x8 g1 = { 0x00020000, (NN << 16), (NN << 16), (40 << 16),
                         24, NN, 0, 0 };
            i32x4 gz4 = { 0, 0, 0, 0 };
            i32x8 gz8 = { 0, 0, 0, 0, 0, 0, 0, 0 };
            __builtin_amdgcn_tensor_load_to_lds(g0, g1, gz4, gz4, gz8, 0);
            __builtin_amdgcn_s_wait_tensorcnt(0);
        }
        return;                                  // caller's __syncthreads publishes LDS
    }
#endif
    for (int i = tid; i < 24*40; i += 512) {
        const int r = i / 40, c = i % 40;
        const int gy = Y0 - 4 + r, gx = X0 - 4 + c;
        float v = 0.0f;
        if (gy >= 0 && gy < NN && gx >= 0 && gx < NN) v = wfc[gy*NN + gx];
        T.sW[r][c] = v;
    }
}

// ---------------- one CPML field update for one 32x16 tile ----------------
__device__ float field_step(Tiles& T, const v2f (&cf)[5],
    const float* __restrict__ wfc, float* wfpn,               // wfpn: read wfp, write wfn
    const float* __restrict__ psiy_in, float* __restrict__ psiy_out,
    const float* __restrict__ psix_in, float* __restrict__ psix_out,
    float* __restrict__ zetay, float* __restrict__ zetax,
    const float* __restrict__ v2dt2,
    const float* __restrict__ pa, const float* __restrict__ pb,
    float pay, float pby, float pax, float pbx,               // profiles at this thread's cell
    const float* __restrict__ born,   // nullptr for background field
    float lap_in, float fsrc, int sy, int sx,
    int X0, int Y0, int tid)
{
    load_tile(T, wfc, X0, Y0, tid);
    __syncthreads();

    // ---- psiy_new = ay*psiy + by*dwdy over rows [-2,17] ----
    for (int i = tid; i < 20*32; i += 512) {
        const int r = i / 32, c = i % 32;
        const int gy = Y0 - 2 + r, gx = X0 + c;
        float val = 0.0f;
        if (gy >= 0 && gy < NN && gx < NN) {
            const float dwdy = INV_H * (C1M2*T.sW[r][c+4]   + C1M1*T.sW[r+1][c+4]
                                      + C1P1*T.sW[r+3][c+4] + C1P2*T.sW[r+4][c+4]);
            val = pa[gy]*psiy_in[gy*NN + gx] + pb[gy]*dwdy;
        }
        T.sPY[r][c] = val;
    }
    // ---- psix_new over cols [-2,33] ----
    for (int i = tid; i < 16*36; i += 512) {
        const int r = i / 36, c = i % 36;
        const int gy = Y0 + r, gx = X0 - 2 + c;
        float val = 0.0f;
        if (gy < NN && gx >= 0 && gx < NN) {
            const float dwdx = INV_H * (C1M2*T.sW[r+4][c]   + C1M1*T.sW[r+4][c+1]
                                      + C1P1*T.sW[r+4][c+3] + C1P2*T.sW[r+4][c+4]);
            val = pa[gx]*psix_in[gy*NN + gx] + pb[gx]*dwdx;
        }
        T.sPX[r][c] = val;
    }
    // ---- second derivatives via f32 WMMA (waves 0..3, wave-uniform branches) ----
    {
        const int wave = tid >> 5, lane = tid & 31;
        if      (wave == 0) d2x_wmma(T.sW, T.sD2X, 0, lane, cf);
        else if (wave == 1) d2x_wmma(T.sW, T.sD2X, 1, lane, cf);
        else if (wave == 2) d2y_wmma(T.sW, T.sD2Y, 0, lane, cf);
        else if (wave == 3) d2y_wmma(T.sW, T.sD2Y, 1, lane, cf);
    }
    __syncthreads();

    // ---- pointwise CPML + leap-frog update ----
    const int tx = threadIdx.x, ty = threadIdx.y;
    const int gy = Y0 + ty, gx = X0 + tx;
    float lap = 0.0f;
    if (gy < NN && gx < NN) {
        const int g = gy*NN + gx;
        const float d2y = T.sD2Y[ty][tx];
        const float d2x = T.sD2X[ty][tx];
        const float dpsiy = INV_H * (C1M2*T.sPY[ty][tx]   + C1M1*T.sPY[ty+1][tx]
                                   + C1P1*T.sPY[ty+3][tx] + C1P2*T.sPY[ty+4][tx]);
        const float dpsix = INV_H * (C1M2*T.sPX[ty][tx]   + C1M1*T.sPX[ty][tx+1]
                                   + C1P1*T.sPX[ty][tx+3] + C1P2*T.sPX[ty][tx+4]);
        const float zy = pay*zetay[g] + pby*(d2y + dpsiy);
        const float zx = pax*zetax[g] + pbx*(d2x + dpsix);
        lap = d2y + d2x + dpsiy + dpsix + zy + zx;
        const float extra = born ? born[g]*lap_in : 0.0f;
        float wfn_v = v2dt2[g]*lap + 2.0f*T.sW[ty+4][tx+4] - wfpn[g] + extra;
        if (gy == sy && gx == sx) wfn_v += fsrc;
        wfpn[g]     = wfn_v;
        psiy_out[g] = T.sPY[ty+2][tx];
        psix_out[g] = T.sPX[ty][tx+2];
        zetay[g]    = zy;
        zetax[g]    = zx;
    }
    return lap;
}

// ---------------- fused per-timestep kernel (both wavefields, all shots) ----------------
__global__ __launch_bounds__(512) void step_kernel(
    const float* __restrict__ wfc, float* wfpn,
    const float* __restrict__ pyi, float* __restrict__ pyo,
    const float* __restrict__ pxi, float* __restrict__ pxo,
    float* __restrict__ zy, float* __restrict__ zx,
    const float* __restrict__ wfcs, float* wfpns,
    const float* __restrict__ pyis, float* __restrict__ pyos,
    const float* __restrict__ pxis, float* __restrict__ pxos,
    float* __restrict__ zys, float* __restrict__ zxs,
    const float* __restrict__ v2dt2, const float* __restrict__ born,
    const float* __restrict__ pa, const float* __restrict__ pb,
    const float* __restrict__ fbg, const float* __restrict__ fsc,
    const int* __restrict__ srcloc, int t)
{
    __shared__ Tiles T;
    const int s = blockIdx.z;
    const size_t off = (size_t)s * NYX;
    const int X0 = blockIdx.x * 32, Y0 = blockIdx.y * 16;
    const int tid = threadIdx.y * 32 + threadIdx.x;
    const int sy = srcloc[s*2 + 0] + PADC;
    const int sx = srcloc[s*2 + 1] + PADC;
    const float f_b = fbg[t*NSH + s];
    const float f_s = fsc[t*NSH + s];

    // per-lane banded-coefficient fragments (shared by d2x-B and d2y-A, both fields)
    v2f cf[5];
    {
        const int lane = tid & 31;
        const int half = lane >> 4, l = lane & 15;
#pragma unroll
        for (int q = 0; q < 5; ++q) {
            const int k0 = 4*q + 2*half;
            cf[q].x = band_c2(k0 - l);
            cf[q].y = band_c2(k0 + 1 - l);
        }
    }
    // hoist per-cell PML profile values (same for both fields)
    const int gy = min(Y0 + (int)threadIdx.y, NN - 1);
    const int gx = min(X0 + (int)threadIdx.x, NN - 1);
    const float pay = pa[gy], pby = pb[gy], pax = pa[gx], pbx = pb[gx];

    __builtin_prefetch(wfpn + off + (size_t)Y0*NN + X0, 0, 3);
    __builtin_prefetch(wfpns + off + (size_t)Y0*NN + X0, 0, 3);

    const float lap_bg = field_step(T, cf, wfc + off, wfpn + off,
                                    pyi + off, pyo + off, pxi + off, pxo + off,
                                    zy + off, zx + off, v2dt2, pa, pb,
                                    pay, pby, pax, pbx,
                                    nullptr, 0.0f, f_b, sy, sx, X0, Y0, tid);
    __syncthreads();  // reuse LDS for scattered field
    field_step(T, cf, wfcs + off, wfpns + off,
               pyis + off, pyos + off, pxis + off, pxos + off,
               zys + off, zxs + off, v2dt2, pa, pb,
               pay, pby, pax, pbx,
               born, lap_bg, f_s, sy, sx, X0, Y0, tid);
}

// ---------------- setup kernels ----------------
__global__ void profiles_kernel(float* pa, float* pb) {
    const int i = blockIdx.x*blockDim.x + threadIdx.x;
    if (i >= NN) return;
    const float f1 = fminf(fmaxf((22.0f - (float)i) / 20.0f, 0.0f), 1.0f);
    const float f2 = fminf(fmaxf(((float)i - 421.0f) / 20.0f, 0.0f), 1.0f);
    const float frac = fmaxf(f1, f2);
    const float smax = 3.0f * 2500.0f * logf(1000.0f) / 200.0f;   // L = PML*DX = 100
    const float sigma = smax * frac * frac;
    const float alpha = 3.14159265358979f * 25.0f;
    const float a = expf(-(sigma + alpha) * DT_C);
    pa[i] = a;
    pb[i] = (sigma > 0.0f) ? sigma / (sigma + alpha) * (a - 1.0f) : 0.0f;
}

__global__ void fields_kernel(const float* __restrict__ v, const float* __restrict__ sc,
                              float* v2dt2, float* born) {
    const int i = blockIdx.x*blockDim.x + threadIdx.x;
    if (i >= NYX) return;
    const int y = i / NN, x = i % NN;
    const int vy = min(max(y - PADC, 0), 399);
    const int vx = min(max(x - PADC, 0), 399);
    const float vp = v[vy*400 + vx];                              // edge pad
    const bool inb = (y >= PADC && y < PADC+400 && x >= PADC && x < PADC+400);
    const float sp = inb ? sc[(y-PADC)*400 + (x-PADC)] : 0.0f;    // zero pad
    v2dt2[i] = vp*DT_C*vp*DT_C;
    born[i]  = 2.0f*vp*sp*DT_C*DT_C;
}

__global__ void famp_kernel(const float* __restrict__ amp, const int* __restrict__ srcloc,
                            const float* __restrict__ v2dt2, const float* __restrict__ born,
                            float* fbg, float* fsc) {
    const int i = blockIdx.x*blockDim.x + threadIdx.x;
    if (i >= NSH*NTT) return;
    const int s = i / NTT, t = i % NTT;
    const int sy = srcloc[s*2+0] + PADC, sx = srcloc[s*2+1] + PADC;
    const int g = sy*NN + sx;
    const float a = amp[s*NTT + t];
    fbg[t*NSH + s] = -a * v2dt2[g];      // -amp * v^2 * dt^2
    fsc[t*NSH + s] = -a * born[g];       // -2*amp * v*sc * dt^2
}

__global__ void record_kernel(const float* __restrict__ wfn_bg, const float* __restrict__ wfn_sc,
                              const int* __restrict__ bgrec, const int* __restrict__ rec,
                              float* out_bg, float* out_sc, int t) {
    const int i = blockIdx.x*blockDim.x + threadIdx.x;
    if (i >= NSH*NRC) return;
    const int s = i / NRC, r = i % NRC;
    {
        const int y = bgrec[(s*NRC + r)*2 + 0] + PADC;
        const int x = bgrec[(s*NRC + r)*2 + 1] + PADC;
        out_bg[((size_t)s*NRC + r)*NTT + t] = wfn_bg[(size_t)s*NYX + y*NN + x];
    }
    {
        const int y = rec[(s*NRC + r)*2 + 0] + PADC;
        const int x = rec[(s*NRC + r)*2 + 1] + PADC;
        out_sc[((size_t)s*NRC + r)*NTT + t] = wfn_sc[(size_t)s*NYX + y*NN + x];
    }
}

__global__ void copy_out_kernel(const float* __restrict__ wfb, const float* __restrict__ wfs,
                                float* out) {
    const size_t i = (size_t)blockIdx.x*blockDim.x + threadIdx.x;
    if (i < (size_t)NSH*NYX) {
        out[i] = wfb[i];
        out[(size_t)NSH*NYX + i] = wfs[i];
    }
}

// ---------------- host driver ----------------
extern "C" void kernel_launch(void* const* d_in, const int* in_sizes, int n_in,
                              void* d_out, int out_size, void* d_ws, size_t ws_size,
                              hipStream_t stream) {
    const float* v        = (const float*)d_in[0];
    const float* scatter  = (const float*)d_in[1];
    const float* amp      = (const float*)d_in[2];
    const int*   srcloc   = (const int*)d_in[3];
    const int*   recloc   = (const int*)d_in[4];
    const int*   bgrecloc = (const int*)d_in[5];
    float* out = (float*)d_out;

    const size_t SHN = (size_t)NSH * NYX;
    float* ws = (float*)d_ws;
    float* wf[2]  = { ws + 0*SHN, ws + 1*SHN };
    float* py[2]  = { ws + 2*SHN, ws + 3*SHN };
    float* px[2]  = { ws + 4*SHN, ws + 5*SHN };
    float* zy     = ws + 6*SHN;
    float* zx     = ws + 7*SHN;
    float* wfs[2] = { ws + 8*SHN, ws + 9*SHN };
    float* pys[2] = { ws + 10*SHN, ws + 11*SHN };
    float* pxs[2] = { ws + 12*SHN, ws + 13*SHN };
    float* zys    = ws + 14*SHN;
    float* zxs    = ws + 15*SHN;
    float* v2dt2  = ws + 16*SHN;
    float* born   = v2dt2 + NYX;
    float* pa     = born + NYX;
    float* pb     = pa + NN;
    float* fbg    = pb + NN;
    float* fsc    = fbg + NSH*NTT;
    const size_t need = (size_t)((fsc + NSH*NTT) - ws) * sizeof(float);
    if (ws_size < need) return;  // workspace too small; nothing safe to do

    // zero all propagation state (initial carry is zeros)
    (void)hipMemsetAsync(d_ws, 0, 16*SHN*sizeof(float), stream);

    profiles_kernel<<<2, 256, 0, stream>>>(pa, pb);
    fields_kernel<<<(NYX + 255)/256, 256, 0, stream>>>(v, scatter, v2dt2, born);
    famp_kernel<<<(NSH*NTT + 255)/256, 256, 0, stream>>>(amp, srcloc, v2dt2, born, fbg, fsc);

    const dim3 blk(32, 16);
    const dim3 grd((NN + 31)/32, (NN + 15)/16, NSH);
    float* orecbg = out + 2*SHN;
    float* orecsc = orecbg + (size_t)NSH*NRC*NTT;

    for (int t = 0; t < NTT; ++t) {
        const int cur = t & 1, nxt = (t + 1) & 1;
        step_kernel<<<grd, blk, 0, stream>>>(
            wf[cur], wf[nxt], py[cur], py[nxt], px[cur], px[nxt], zy, zx,
            wfs[cur], wfs[nxt], pys[cur], pys[nxt], pxs[cur], pxs[nxt], zys, zxs,
            v2dt2, born, pa, pb, fbg, fsc, srcloc, t);
        record_kernel<<<2, 256, 0, stream>>>(wf[nxt], wfs[nxt], bgrecloc, recloc,
                                             orecbg, orecsc, t);
    }
    // after t=299 the latest wfn lives in buffer index (300 & 1) == 0
    copy_out_kernel<<<(int)((SHN + 255)/256), 256, 0, stream>>>(wf[0], wfs[0], out);
}